// OUR_CUSTOM_SSIM_PYRAMID_52767968199432
// MI455X (gfx1250) — compile-verified
//
#include <hip/hip_runtime.h>

typedef __attribute__((ext_vector_type(2))) float v2f;
typedef __attribute__((ext_vector_type(8))) float v8f;

// Banded 0/1 matrix generator for the 11-tap box filter:
// band01(m, r) == 1 iff input index r lies in [m, m+10] and r < 26 (26 = 16+10 valid taps).
__device__ __forceinline__ float band01(int m, int r) {
    unsigned d = (unsigned)(r - m);
    return (d <= 10u && r < 26) ? 1.0f : 0.0f;
}

__device__ __forceinline__ v8f wmma4(v2f a, v2f b, v8f c) {
    // V_WMMA_F32_16X16X4_F32 : D(16x16,f32) = A(16x4,f32) * B(4x16,f32) + C
    return __builtin_amdgcn_wmma_f32_16x16x4_f32(
        /*neg_a=*/false, a, /*neg_b=*/false, b,
        /*c_mod=*/(short)0, c, /*reuse_a=*/false, /*reuse_b=*/false);
}

// One wave (32 lanes) computes one 16x16 output tile.
// Vertical 11-tap box sum of 5 channels (p1, p2, p1^2, p2^2, p1*p2) via WMMA
// (banded A from lane IDs, image data as B), transpose through per-wave LDS,
// horizontal 11-tap box sum via WMMA, then closed-form SSIM finalize + block reduce.
#define LROW 33   // padded colsum row stride (33 coprime to 64 banks -> conflict-free)
#define LWAVE 544 // 16 * 33 rounded up

__global__ __launch_bounds__(256) void ssim_level_kernel(
    const float* __restrict__ img1, const float* __restrict__ img2,
    int H, int W, float scale, float* __restrict__ partials)
{
    __shared__ float cbuf[8 * LWAVE];

    const int lane = threadIdx.x & 31;
    const int wave = threadIdx.x >> 5;

    const int tilesX = W >> 4;
    const int tilesY = H >> 4;
    int gw  = blockIdx.x * 8 + wave;
    int tx  = gw % tilesX;
    int tmp = gw / tilesX;
    int ty  = tmp % tilesY;
    int b   = tmp / tilesY;
    const int x0 = tx << 4;
    const int y0 = ty << 4;

    const float* i1 = img1 + (size_t)b * H * W;
    const float* i2 = img2 + (size_t)b * H * W;

    const int n   = lane & 15;          // A: M row / B: N col (lanes 0-15 == 16-31)
    const int khi = (lane >> 4) << 1;   // lanes 16-31 hold K = 2,3 of each chunk

    // Band-matrix A values, shared by vertical and horizontal passes.
    float A0[7], A1[7];
    #pragma unroll
    for (int kc = 0; kc < 7; ++kc) {
        A0[kc] = band01(n, 4*kc + khi + 0);
        A1[kc] = band01(n, 4*kc + khi + 1);
    }

    // Column clamp + 0/1 mask, one per 16-wide half (invariant over K-chunks).
    int col_c[2];
    float cmask[2];
    #pragma unroll
    for (int h = 0; h < 2; ++h) {
        int c = x0 - 5 + 16*h + n;
        cmask[h] = ((unsigned)c < (unsigned)W) ? 1.0f : 0.0f;
        col_c[h] = c < 0 ? 0 : (c >= W ? W - 1 : c);
    }

    // ---- Vertical pass: colsum[16 rows][32 cols] for 5 channels -------------
    v8f Cv[5][2];
    #pragma unroll
    for (int c = 0; c < 5; ++c)
        #pragma unroll
        for (int h = 0; h < 2; ++h)
            Cv[c][h] = (v8f){0,0,0,0,0,0,0,0};

    #pragma unroll
    for (int kc = 0; kc < 7; ++kc) {
        v2f A = {A0[kc], A1[kc]};
        int ya = y0 - 5 + 4*kc + khi;
        int yb = ya + 1;
        float mya = ((unsigned)ya < (unsigned)H) ? 1.0f : 0.0f;
        float myb = ((unsigned)yb < (unsigned)H) ? 1.0f : 0.0f;
        int ya_c = ya < 0 ? 0 : (ya >= H ? H - 1 : ya);
        int yb_c = yb < 0 ? 0 : (yb >= H ? H - 1 : yb);
        int rowa = ya_c * W;
        int rowb = yb_c * W;
        #pragma unroll
        for (int h = 0; h < 2; ++h) {
            float fa = mya * cmask[h];
            float fb = myb * cmask[h];
            int oa = rowa + col_c[h];
            int ob = rowb + col_c[h];
            // Unconditional loads from clamped (always valid) addresses,
            // zeroed by mask multiply for taps outside the image.
            float p1a = i1[oa] * fa;
            float p2a = i2[oa] * fa;
            float p1b = i1[ob] * fb;
            float p2b = i2[ob] * fb;
            v2f B0 = {p1a,       p1b};
            v2f B1 = {p2a,       p2b};
            v2f B2 = {p1a * p1a, p1b * p1b};
            v2f B3 = {p2a * p2a, p2b * p2b};
            v2f B4 = {p1a * p2a, p1b * p2b};
            Cv[0][h] = wmma4(A, B0, Cv[0][h]);
            Cv[1][h] = wmma4(A, B1, Cv[1][h]);
            Cv[2][h] = wmma4(A, B2, Cv[2][h]);
            Cv[3][h] = wmma4(A, B3, Cv[3][h]);
            Cv[4][h] = wmma4(A, B4, Cv[4][h]);
        }
    }

    // ---- Transpose (per-wave LDS, in-order DS ops) + horizontal pass --------
    float* buf = cbuf + wave * LWAVE;
    v8f F[5];
    #pragma unroll
    for (int c = 0; c < 5; ++c) {
        // C layout: VGPR r, lanes 0-15 -> M=r, lanes 16-31 -> M=r+8; N = 16h + n
        #pragma unroll
        for (int h = 0; h < 2; ++h)
            #pragma unroll
            for (int rv = 0; rv < 8; ++rv) {
                int M  = rv + ((lane >> 4) << 3);
                int Nc = 16*h + n;
                buf[M*LROW + Nc] = Cv[c][h][rv];
            }
        v8f acc = (v8f){0,0,0,0,0,0,0,0};
        #pragma unroll
        for (int kc = 0; kc < 7; ++kc) {
            v2f A = {A0[kc], A1[kc]};           // A = Mh^T band (same generator)
            int r0 = 4*kc + khi;                // B[K][m] = colsum[m][K]
            v2f Bv = { buf[n*LROW + r0], buf[n*LROW + r0 + 1] };
            acc = wmma4(A, Bv, acc);
        }
        F[c] = acc;
    }

    // ---- Finalize: val = sig1/s1^2 + sig2/s2^2 - 2 cov/(s1 s2) --------------
    const float inv121 = 1.0f / 121.0f;
    const float eps  = 1e-6f;
    const float eps2 = eps * eps;
    float acc = 0.0f;
    #pragma unroll
    for (int rv = 0; rv < 8; ++rv) {
        float S1  = F[0][rv], S2  = F[1][rv];
        float S11 = F[2][rv], S22 = F[3][rv], S12 = F[4][rv];
        float mu1 = S1 * inv121, mu2 = S2 * inv121;
        float sig1 = S11 * inv121 - mu1 * mu1;
        float sig2 = S22 * inv121 - mu2 * mu2;
        float cov  = S12 * inv121 - mu1 * mu2;
        float s1 = sqrtf(fmaxf(sig1, 0.0f) + eps2) + eps;
        float s2 = sqrtf(fmaxf(sig2, 0.0f) + eps2) + eps;
        acc += sig1 / (s1 * s1) + sig2 / (s2 * s2) - 2.0f * cov / (s1 * s2);
    }
    acc *= scale;

    // ---- Deterministic block reduction -> per-block partial -----------------
    __syncthreads();
    cbuf[threadIdx.x] = acc;
    __syncthreads();
    #pragma unroll
    for (int s = 128; s > 0; s >>= 1) {
        if ((int)threadIdx.x < s) cbuf[threadIdx.x] += cbuf[threadIdx.x + s];
        __syncthreads();
    }
    if (threadIdx.x == 0) partials[blockIdx.x] = cbuf[0];
}

// torch-style bicubic (a=-0.75) exact-half downsample: fixed separable 4-tap,
// edge-replicated (index clamped) taps.
__global__ __launch_bounds__(256) void down_kernel(
    const float* __restrict__ in, float* __restrict__ out, int B, int H, int W)
{
    const int H2 = H >> 1, W2 = W >> 1;
    long idx = (long)blockIdx.x * blockDim.x + threadIdx.x;
    long total = (long)B * H2 * W2;
    if (idx >= total) return;
    int x = (int)(idx % W2);
    long t = idx / W2;
    int y = (int)(t % H2);
    int b = (int)(t / H2);
    const float k1d[4] = {-0.09375f, 0.59375f, 0.59375f, -0.09375f};
    const float* p = in + (long)b * H * W;
    float sum = 0.0f;
    #pragma unroll
    for (int i = 0; i < 4; ++i) {
        int yy = 2*y - 1 + i; yy = yy < 0 ? 0 : (yy >= H ? H - 1 : yy);
        int row = yy * W;
        float rowv = 0.0f;
        #pragma unroll
        for (int j = 0; j < 4; ++j) {
            int xx = 2*x - 1 + j; xx = xx < 0 ? 0 : (xx >= W ? W - 1 : xx);
            rowv += k1d[j] * p[row + xx];
        }
        sum += k1d[i] * rowv;
    }
    out[idx] = sum;
}

__global__ __launch_bounds__(256) void reduce_kernel(
    const float* __restrict__ partials, int nTot, float* __restrict__ out)
{
    __shared__ float red[256];
    float a = 0.0f;
    for (int i = threadIdx.x; i < nTot; i += 256) a += partials[i];
    red[threadIdx.x] = a;
    __syncthreads();
    #pragma unroll
    for (int s = 128; s > 0; s >>= 1) {
        if ((int)threadIdx.x < s) red[threadIdx.x] += red[threadIdx.x + s];
        __syncthreads();
    }
    if (threadIdx.x == 0) out[0] = red[0];
}

extern "C" void kernel_launch(void* const* d_in, const int* in_sizes, int n_in,
                              void* d_out, int out_size, void* d_ws, size_t ws_size,
                              hipStream_t stream)
{
    const float* img1 = (const float*)d_in[0];
    const float* img2 = (const float*)d_in[1];
    float* out = (float*)d_out;
    float* ws  = (float*)d_ws;

    // d_ws layout (floats):
    float* partials = ws;                      // [0, 1024): 512 + 128 + 32 used
    float* a1 = ws + 1024;                     // 4*256*256 level-1 img1
    float* b1 = a1 + 4 * 256 * 256;            // 4*256*256 level-1 img2
    float* a2 = b1 + 4 * 256 * 256;            // 4*128*128 level-2 img1
    float* b2 = a2 + 4 * 128 * 128;            // 4*128*128 level-2 img2

    const int B = 4;

    // Level 0: 512x512, weight 0.2  (32*32*4 = 4096 tiles / 8 waves = 512 blocks)
    ssim_level_kernel<<<512, 256, 0, stream>>>(img1, img2, 512, 512,
                                               0.2f / (4.0f * 512.0f * 512.0f),
                                               partials + 0);
    down_kernel<<<(B*256*256 + 255)/256, 256, 0, stream>>>(img1, a1, B, 512, 512);
    down_kernel<<<(B*256*256 + 255)/256, 256, 0, stream>>>(img2, b1, B, 512, 512);

    // Level 1: 256x256, weight 0.4  (1024 tiles -> 128 blocks)
    ssim_level_kernel<<<128, 256, 0, stream>>>(a1, b1, 256, 256,
                                               0.4f / (4.0f * 256.0f * 256.0f),
                                               partials + 512);
    down_kernel<<<(B*128*128 + 255)/256, 256, 0, stream>>>(a1, a2, B, 256, 256);
    down_kernel<<<(B*128*128 + 255)/256, 256, 0, stream>>>(b1, b2, B, 256, 256);

    // Level 2: 128x128, weight 0.6  (256 tiles -> 32 blocks)
    ssim_level_kernel<<<32, 256, 0, stream>>>(a2, b2, 128, 128,
                                              0.6f / (4.0f * 128.0f * 128.0f),
                                              partials + 640);

    // Fixed-order final reduction -> scalar
    reduce_kernel<<<1, 256, 0, stream>>>(partials, 672, out);
}